// MessageBlock_75823352644259
// MI455X (gfx1250) — compile-verified
//
#include <hip/hip_runtime.h>

#define N_NODES 20000
#define N_EDGES 320000
#define EMB     128
#define EMB3    384
#define N_RBF   20
#define R_CUT   5.0f
#define PI_F    3.14159265358979323846f
#define EB      64        // edges per block in edge kernel

typedef float v2f __attribute__((ext_vector_type(2)));
typedef float v8f __attribute__((ext_vector_type(8)));

// Guaranteed-native f32 scatter-add: device-scope atomic handled at L2.
// (atomic-without-return -> STOREcnt; S_ENDPGM implicitly waits idle.)
__device__ __forceinline__ void atom_add_f32(float* p, float v) {
  asm volatile("global_atomic_add_f32 %0, %1, off scope:SCOPE_DEV"
               :
               : "v"(p), "v"(v)
               : "memory");
}

// ---------------------------------------------------------------------------
// Kernel 1: initialize outputs with the residual base: s_out = s, v_out = v.
// ---------------------------------------------------------------------------
__global__ __launch_bounds__(256) void init_out(const float* __restrict__ s,
                                                const float* __restrict__ v,
                                                float* __restrict__ out) {
  const int S4 = (N_NODES * EMB) / 4;
  const int T4 = (N_NODES * EMB + N_NODES * 3 * EMB) / 4;
  int i = blockIdx.x * blockDim.x + threadIdx.x;
  if (i < T4) {
    float4 val = (i < S4) ? ((const float4*)s)[i] : ((const float4*)v)[i - S4];
    ((float4*)out)[i] = val;
  }
}

// ---------------------------------------------------------------------------
// Kernel 2: node MLP  s_pass = silu(s@W1 + b1) @ W2 + b2   -> ws [N, 384]
// One block = 16 nodes (one WMMA M-tile), 128 threads = 4 waves.
// V_WMMA_F32_16X16X4_F32, K-loop of 32 steps covers EMB=128.
// ---------------------------------------------------------------------------
__global__ __launch_bounds__(128) void node_mlp(const float* __restrict__ s,
                                                const float* __restrict__ W1,
                                                const float* __restrict__ b1,
                                                const float* __restrict__ W2,
                                                const float* __restrict__ b2,
                                                float* __restrict__ s_pass) {
  __shared__ float lds_s[16 * EMB];
  __shared__ float lds_h[16 * EMB];

  const int tid  = threadIdx.x;
  const int lane = tid & 31;
  const int wave = tid >> 5;
  const int node0 = blockIdx.x * 16;

  const int halfsel = lane >> 4;   // 0: lanes 0-15, 1: lanes 16-31
  const int lmod    = lane & 15;

  for (int i = tid; i < 16 * EMB; i += 128)
    lds_s[i] = s[(size_t)node0 * EMB + i];
  __syncthreads();

  // ---- Phase 1: h = silu(s @ W1 + b1); 8 col-tiles, 2 per wave ----
  for (int t = 0; t < 2; ++t) {
    const int colbase = (wave * 2 + t) * 16;
    v8f acc = {};
    for (int k0 = 0; k0 < EMB; k0 += 4) {
      const int ka = k0 + 2 * halfsel;
      v2f a, b;
      a.x = lds_s[lmod * EMB + ka];
      a.y = lds_s[lmod * EMB + ka + 1];
      b.x = W1[(size_t)ka * EMB + colbase + lmod];
      b.y = W1[(size_t)(ka + 1) * EMB + colbase + lmod];
      acc = __builtin_amdgcn_wmma_f32_16x16x4_f32(
          false, a, false, b, (short)0, acc, false, false);
    }
    const int col  = colbase + lmod;
    const float bias = b1[col];
#pragma unroll
    for (int i = 0; i < 8; ++i) {
      const int row = i + 8 * halfsel;
      float x = acc[i] + bias;
      lds_h[row * EMB + col] = x / (1.0f + __expf(-x));   // SiLU
    }
  }
  __syncthreads();

  // ---- Phase 2: s_pass = h @ W2 + b2; 24 col-tiles, 6 per wave ----
  for (int t = 0; t < 6; ++t) {
    const int colbase = (wave * 6 + t) * 16;
    v8f acc = {};
    for (int k0 = 0; k0 < EMB; k0 += 4) {
      const int ka = k0 + 2 * halfsel;
      v2f a, b;
      a.x = lds_h[lmod * EMB + ka];
      a.y = lds_h[lmod * EMB + ka + 1];
      b.x = W2[(size_t)ka * EMB3 + colbase + lmod];
      b.y = W2[(size_t)(ka + 1) * EMB3 + colbase + lmod];
      acc = __builtin_amdgcn_wmma_f32_16x16x4_f32(
          false, a, false, b, (short)0, acc, false, false);
    }
    const int col  = colbase + lmod;
    const float bias = b2[col];
#pragma unroll
    for (int i = 0; i < 8; ++i) {
      const int row = i + 8 * halfsel;
      s_pass[(size_t)(node0 + row) * EMB3 + col] = acc[i] + bias;
    }
  }
}

// ---------------------------------------------------------------------------
// Kernel 3: per-edge message + scatter. Block = 64 edges, thread = column.
// Each thread keeps its 20x3 Wrbf column slice in registers (reused across
// all 64 edges -> Wrbf L2 traffic drops 64x). Per-edge scalars (rbf values,
// fcut, indices, r_hat) staged in LDS once per block, broadcast-read.
// ---------------------------------------------------------------------------
__global__ __launch_bounds__(128) void edge_msg(const float* __restrict__ s_pass,
                                                const float* __restrict__ v,
                                                const int*   __restrict__ edges,
                                                const float* __restrict__ r_ij,
                                                const float* __restrict__ rhat,
                                                const float* __restrict__ Wrbf,
                                                const float* __restrict__ brbf,
                                                float* __restrict__ s_out,
                                                float* __restrict__ v_out) {
  const int ebase = blockIdx.x * EB;
  const int c = threadIdx.x;                 // column 0..127

  __shared__ float rbf_sh[EB][N_RBF];        // 5 KB
  __shared__ float fc_sh[EB];
  __shared__ int   dst_sh[EB], src_sh[EB];
  __shared__ float rn_sh[EB][3];

  // thread-private Wrbf column slice: w[k][third]
  float wcol[N_RBF][3];
#pragma unroll
  for (int k = 0; k < N_RBF; ++k) {
    const float* wr = Wrbf + (size_t)k * EMB3;
    wcol[k][0] = wr[c];
    wcol[k][1] = wr[EMB + c];
    wcol[k][2] = wr[2 * EMB + c];
  }
  const float bc0 = brbf[c], bc1 = brbf[EMB + c], bc2 = brbf[2 * EMB + c];

  // cooperative staging of per-edge scalars
  for (int i = c; i < EB * N_RBF; i += 128) {          // 10 per thread
    const int e = i / N_RBF, k = i % N_RBF;
    const float r = r_ij[ebase + e];
    rbf_sh[e][k] = __sinf((float)(k + 1) * (PI_F / R_CUT) * r) / r;
  }
  if (c < EB) {
    const float r = r_ij[ebase + c];
    fc_sh[c]  = (r < R_CUT) ? 0.5f * (__cosf(PI_F * r / R_CUT) + 1.0f) : 0.0f;
    dst_sh[c] = edges[2 * (ebase + c) + 0];
    src_sh[c] = edges[2 * (ebase + c) + 1];
    rn_sh[c][0] = rhat[3 * (ebase + c) + 0];
    rn_sh[c][1] = rhat[3 * (ebase + c) + 1];
    rn_sh[c][2] = rhat[3 * (ebase + c) + 2];
  }
  __syncthreads();

  for (int ee = 0; ee < EB; ++ee) {
    const int dst = dst_sh[ee];
    const int src = src_sh[ee];

    float a0 = bc0, a1 = bc1, a2 = bc2;
#pragma unroll
    for (int k = 0; k < N_RBF; ++k) {
      const float rv = rbf_sh[ee][k];        // LDS broadcast
      a0 = fmaf(rv, wcol[k][0], a0);
      a1 = fmaf(rv, wcol[k][1], a1);
      a2 = fmaf(rv, wcol[k][2], a2);
    }
    const float fc = fc_sh[ee];
    a0 = a0 * a0 * fc;  a1 = a1 * a1 * fc;  a2 = a2 * a2 * fc;

    const float* sp = s_pass + (size_t)src * EMB3;
    const float dv = a0 * sp[c];             // delta_v gate
    const float ds = a1 * sp[EMB + c];       // delta_s
    const float dr = a2 * sp[2 * EMB + c];   // delta_rep

    atom_add_f32(&s_out[(size_t)dst * EMB + c], ds);

    const float* vs = v + (size_t)src * 3 * EMB;
    float* vo = v_out + (size_t)dst * 3 * EMB;
    const float rn0 = rn_sh[ee][0], rn1 = rn_sh[ee][1], rn2 = rn_sh[ee][2];
    atom_add_f32(&vo[c],           fmaf(vs[c],           dv, rn0 * dr));
    atom_add_f32(&vo[EMB + c],     fmaf(vs[EMB + c],     dv, rn1 * dr));
    atom_add_f32(&vo[2 * EMB + c], fmaf(vs[2 * EMB + c], dv, rn2 * dr));
  }
}

// ---------------------------------------------------------------------------
extern "C" void kernel_launch(void* const* d_in, const int* in_sizes, int n_in,
                              void* d_out, int out_size, void* d_ws, size_t ws_size,
                              hipStream_t stream) {
  const float* s     = (const float*)d_in[0];
  const float* v     = (const float*)d_in[1];
  const int*   edges = (const int*)  d_in[2];
  const float* r_ij  = (const float*)d_in[3];
  const float* rhat  = (const float*)d_in[4];
  const float* W1    = (const float*)d_in[5];
  const float* b1    = (const float*)d_in[6];
  const float* W2    = (const float*)d_in[7];
  const float* b2    = (const float*)d_in[8];
  const float* Wrbf  = (const float*)d_in[9];
  const float* brbf  = (const float*)d_in[10];

  float* s_out  = (float*)d_out;
  float* v_out  = s_out + (size_t)N_NODES * EMB;
  float* s_pass = (float*)d_ws;   // [N_NODES, 384] f32 = 30.72 MB scratch

  const int total4 = (N_NODES * EMB + N_NODES * 3 * EMB) / 4;
  init_out<<<(total4 + 255) / 256, 256, 0, stream>>>(s, v, (float*)d_out);
  node_mlp<<<N_NODES / 16, 128, 0, stream>>>(s, W1, b1, W2, b2, s_pass);
  edge_msg<<<N_EDGES / EB, 128, 0, stream>>>(s_pass, v, edges, r_ij, rhat,
                                             Wrbf, brbf, s_out, v_out);
}